// SeasonalEnvelopeAdapter_3882650437177
// MI455X (gfx1250) — compile-verified
//
#include <hip/hip_runtime.h>
#include <hip/hip_bf16.h>
#include <cmath>

// SeasonalEnvelopeAdapter for gfx1250 (MI455X).
//
// Math: editing 4 rFFT bins is a rank-8 linear correction:
//   y_corr = y_hat + (2/T) * sum_k Re((s_kv-1) * c_k * e^{i w_k t})
//   c_k[b,v] = sum_t y_hat[b,t,v] * (cos(w_k t) - i sin(w_k t))
// Pass 1: projection P[b,v,j] via V_WMMA_F32_16X16X4_F32 (A = Y_b^T tile,
//         B = cos/sin basis from LDS), then fold in the complex gain ->
//         E[b,v,8] coefficients (4 MB workspace, L2-resident).
// Pass 2: memory-bound rank-8 reconstruction + piecewise-linear envelope,
//         fully coalesced along the contiguous V=64 axis.
// HBM traffic ~1.13 GB -> ~48 us at 23.3 TB/s; 3 GFLOP of math is noise.

#define BB   2048
#define TT   720
#define VV   64
#define KK   4
#define PHI_MAX 0.25f
#define TWO_PI_OVER_T 0.008726646259971648f   // 2*pi/720

typedef __attribute__((ext_vector_type(2))) float v2f;
typedef __attribute__((ext_vector_type(8))) float v8f;

#define BAS_STRIDE 724   // 720 padded; 724 mod 64 = 20 -> conflict-free lanes

__device__ __forceinline__ float satf(float x) {
    return fminf(fmaxf(x, 0.0f), 1.0f);
}

// ---------------------------------------------------------------------------
// Pass 1: per batch row b, compute E[b, v, 8].
// Block = 128 threads = 4 waves; wave w owns v-tile [16w, 16w+16).
// GEMM: P[16v x 16n] += A[16v x 4t] * B[4t x 16n], 180 k-steps of
// v_wmma_f32_16x16x4_f32. Only n<8 columns are live (cos/sin pairs).
// ---------------------------------------------------------------------------
__global__ void __launch_bounds__(128)
sea_pass1_project(const float* __restrict__ y,
                  const float* __restrict__ a,
                  const float* __restrict__ phi,
                  const int*   __restrict__ kb,
                  float*       __restrict__ E) {
    __shared__ float sBasT[16 * BAS_STRIDE];   // basis^T: [n][t], rows 8..15 zero
    __shared__ float sProj[4 * 16 * 16];       // per-wave P fragment dump

    const int tid = threadIdx.x;
    const int b   = blockIdx.x;

    // Build transposed basis table: sBasT[n][t] ; n=2k -> cos, n=2k+1 -> sin.
    // Exact integer phase reduction keeps fp32 trig accurate at large t.
    for (int i = tid; i < 16 * BAS_STRIDE; i += 128) {
        const int n = i / BAS_STRIDE;
        const int t = i - n * BAS_STRIDE;
        float val = 0.0f;
        if (n < 8 && t < TT) {
            const int k = n >> 1;
            const int r = (kb[k] * t) % TT;
            const float ang = (float)r * TWO_PI_OVER_T;
            val = (n & 1) ? sinf(ang) : cosf(ang);
        }
        sBasT[i] = val;
    }
    __syncthreads();

    const int wave = tid >> 5;
    const int lane = tid & 31;
    const int m    = lane & 15;          // A row (v within tile) / B,C column
    const int koff = (lane >> 4) << 1;   // lanes 0-15: K=0,1 ; lanes 16-31: K=2,3
    const int v0   = wave * 16;

    // A element (M=m, K=t): Y_b^T[m][t] = y[(b*T + t)*V + v0 + m]
    const float* yA = y + ((size_t)b * TT + koff) * VV + v0 + m;
    // B element (K=t, N=m): sBasT[m][t]
    const float* bB = sBasT + m * BAS_STRIDE + koff;

    v8f c = {0.f, 0.f, 0.f, 0.f, 0.f, 0.f, 0.f, 0.f};
    #pragma unroll 4
    for (int t0 = 0; t0 < TT; t0 += 4) {
        v2f av, bv;
        av.x = yA[t0 * VV];           // K = t0+koff      (64B-coalesced per half-wave)
        av.y = yA[t0 * VV + VV];      // K = t0+koff+1
        const float2 bp = *(const float2*)(bB + t0);  // contiguous pair in LDS
        bv.x = bp.x;
        bv.y = bp.y;
        c = __builtin_amdgcn_wmma_f32_16x16x4_f32(
                false, av, false, bv, (short)0, c, false, false);
    }

    // Dump fragment: lane<16 -> (M=j, N=lane); lane>=16 -> (M=j+8, N=lane-16).
    const int mBase = (lane < 16) ? 0 : 8;
    #pragma unroll
    for (int j = 0; j < 8; ++j)
        sProj[wave * 256 + (mBase + j) * 16 + m] = c[j];
    __syncthreads();

    // Fold complex gain (s-1) into reconstruction coefficients E[b,v,0..7]:
    //   c_k = P0 - i*P1 ;  g = (1+a)e^{i*tanh(phi)*PHI_MAX} - 1 ;  d = g*c
    //   corr(t) = (2/T)(dRe cos - dIm sin) = e0*cos + e1*sin
    for (int q = tid; q < VV * KK; q += 128) {
        const int v = q >> 2, k = q & 3;
        const int wv = v >> 4, mv = v & 15;
        const float P0 = sProj[wv * 256 + mv * 16 + 2 * k];
        const float P1 = sProj[wv * 256 + mv * 16 + 2 * k + 1];
        const float amp = 1.0f + a[v * KK + k];
        const float ph  = tanhf(phi[v * KK + k]) * PHI_MAX;
        const float gRe = amp * cosf(ph) - 1.0f;
        const float gIm = amp * sinf(ph);
        const float dRe = gRe * P0 + gIm * P1;
        const float dIm = gIm * P0 - gRe * P1;
        const float s2  = 2.0f / (float)TT;
        float2 ev;
        ev.x =  s2 * dRe;
        ev.y = -s2 * dIm;
        *(float2*)(E + ((size_t)b * VV + v) * 8 + 2 * k) = ev;
    }
}

// ---------------------------------------------------------------------------
// Pass 2: y_out[b,t,v] = env(v,t) * (y_hat[b,t,v] + sum_j E[b,v,j]*bas[t,j]).
// Grid (B, 4): block owns (b, 180-t chunk); thread owns one v, strides t.
// All global accesses coalesced along v (contiguous 1 KB per step).
// ---------------------------------------------------------------------------
__global__ void __launch_bounds__(256)
sea_pass2_reconstruct(const float* __restrict__ y,
                      const float* __restrict__ knots,
                      const int*   __restrict__ kb,
                      const float* __restrict__ E,
                      float*       __restrict__ out) {
    __shared__ float sBas[180 * 8];   // [t_local][j]: cos,sin pairs for 4 bins

    const int tid   = threadIdx.x;
    const int b     = blockIdx.x;
    const int tbase = blockIdx.y * 180;

    for (int i = tid; i < 180 * 8; i += 256) {
        const int tl = i >> 3, j = i & 7;
        const int k  = j >> 1;
        const int r  = (kb[k] * (tbase + tl)) % TT;
        const float ang = (float)r * TWO_PI_OVER_T;
        sBas[i] = (j & 1) ? sinf(ang) : cosf(ang);
    }
    __syncthreads();

    const int v  = tid & 63;
    const int ts = tid >> 6;   // 4 t-slices so loads cover 4 consecutive t rows

    const size_t eo = ((size_t)b * VV + v) * 8;
    const float4 eL = *(const float4*)(E + eo);
    const float4 eH = *(const float4*)(E + eo + 4);

    // Envelope knots, clipped to [0.5, 1.5]; branchless piecewise-linear.
    const float k0 = fminf(fmaxf(knots[v * 4 + 0], 0.5f), 1.5f);
    const float k1 = fminf(fmaxf(knots[v * 4 + 1], 0.5f), 1.5f);
    const float k2 = fminf(fmaxf(knots[v * 4 + 2], 0.5f), 1.5f);
    const float k3 = fminf(fmaxf(knots[v * 4 + 3], 0.5f), 1.5f);

    const float* yrow = y   + ((size_t)b * TT + tbase) * VV + v;
    float*       orow = out + ((size_t)b * TT + tbase) * VV + v;

    #pragma unroll 3
    for (int i = 0; i < 45; ++i) {
        const int tl = 4 * i + ts;
        const float yv = yrow[(size_t)tl * VV];
        const float4 bL = *(const float4*)(sBas + tl * 8);
        const float4 bH = *(const float4*)(sBas + tl * 8 + 4);
        const float corr = eL.x * bL.x + eL.y * bL.y + eL.z * bL.z + eL.w * bL.w
                         + eH.x * bH.x + eH.y * bH.y + eH.z * bH.z + eH.w * bH.w;
        const int t = tbase + tl;
        const float pos = (float)t * (3.0f / 719.0f);   // linspace(0,1,T)*(K-1)
        const float env = k0 + (k1 - k0) * satf(pos)
                             + (k2 - k1) * satf(pos - 1.0f)
                             + (k3 - k2) * satf(pos - 2.0f);
        orow[(size_t)tl * VV] = env * (yv + corr);
    }
}

extern "C" void kernel_launch(void* const* d_in, const int* in_sizes, int n_in,
                              void* d_out, int out_size, void* d_ws, size_t ws_size,
                              hipStream_t stream) {
    const float* y_hat     = (const float*)d_in[0];  // [B,T,V] f32
    const float* a         = (const float*)d_in[1];  // [V,K]   f32
    const float* phi       = (const float*)d_in[2];  // [V,K]   f32
    const float* env_knots = (const float*)d_in[3];  // [V,4]   f32
    const int*   k_bins    = (const int*)d_in[4];    // [K]     i32
    float*       out       = (float*)d_out;          // [B,T,V] f32
    float*       E         = (float*)d_ws;           // [B,V,8] f32 = 4 MB

    sea_pass1_project<<<dim3(BB), dim3(128), 0, stream>>>(y_hat, a, phi, k_bins, E);
    sea_pass2_reconstruct<<<dim3(BB, 4), dim3(256), 0, stream>>>(y_hat, env_knots,
                                                                 k_bins, E, out);
    (void)in_sizes; (void)n_in; (void)out_size; (void)ws_size;
}